// GCN_75935021794064
// MI455X (gfx1250) — compile-verified
//
#include <hip/hip_runtime.h>
#include <math.h>

typedef __attribute__((ext_vector_type(2))) float v2f;
typedef __attribute__((ext_vector_type(8))) float v8f;

// ---------------- degree / normalization ----------------

__global__ void deg_count_kernel(const long long* __restrict__ dst,
                                 unsigned* __restrict__ deg, long long E) {
  long long e = (long long)blockIdx.x * blockDim.x + threadIdx.x;
  if (e < E) atomicAdd(&deg[(int)dst[e]], 1u);
}

__global__ void dinv_kernel(const unsigned* __restrict__ deg,
                            float* __restrict__ dinv, int N) {
  int i = blockIdx.x * blockDim.x + threadIdx.x;
  if (i < N) dinv[i] = rsqrtf((float)deg[i] + 1.0f);
}

// ---------------- fp32 WMMA GEMM: C[M,64] = A[M,K] @ B[K,64] ----------------
// One wave computes a full 16x64 output row-block: 4 independent 16x16
// accumulators over V_WMMA_F32_16X16X4_F32. The A fragment (1x b64 load per
// k-step) is reused by all 4 WMMAs; the 4 WMMAs chain independent
// accumulators so they issue back-to-back with no D->A/B RAW hazard.
// A layout (16x4, 2 VGPR): lanes 0-15 -> M=lane, K={0,1}; lanes 16-31 -> M=lane-16, K={2,3}
// B layout (4x16, 2 VGPR): VGPR0 rows {0,2}, VGPR1 rows {1,3}; col = lane&15
// C/D (16x16, 8 VGPR): VGPR r, lane L -> M = r + (L>=16 ? 8 : 0), N = L&15
__global__ void __launch_bounds__(256)
gemm_wmma_f32(const float* __restrict__ A, const float* __restrict__ B,
              float* __restrict__ C, int M, int K, int N /* must be 64 */) {
  const int wave = threadIdx.x >> 5;
  const int lane = threadIdx.x & 31;
  long long tileM = (long long)blockIdx.x * 8 + wave;
  if (tileM * 16 >= M) return;

  const int half = lane >> 4;   // 0 or 1
  const int mn   = lane & 15;   // A row / B col / C col
  const int koff = half << 1;   // 0 or 2

  const float* arow = A + ((long long)(tileM * 16 + mn)) * K + koff;
  const float* brow = B + mn;

  v8f c0 = {}, c1 = {}, c2 = {}, c3 = {};
#pragma unroll 2
  for (int k = 0; k < K; k += 4) {
    float2 av = *reinterpret_cast<const float2*>(arow + k);
    v2f a; a.x = av.x; a.y = av.y;
    const float* b0 = brow + (k + koff) * N;
    const float* b1 = brow + (k + koff + 1) * N;
    v2f b;
    b.x = b0[0];  b.y = b1[0];
    c0 = __builtin_amdgcn_wmma_f32_16x16x4_f32(false, a, false, b, (short)0, c0, false, false);
    b.x = b0[16]; b.y = b1[16];
    c1 = __builtin_amdgcn_wmma_f32_16x16x4_f32(false, a, false, b, (short)0, c1, false, false);
    b.x = b0[32]; b.y = b1[32];
    c2 = __builtin_amdgcn_wmma_f32_16x16x4_f32(false, a, false, b, (short)0, c2, false, false);
    b.x = b0[48]; b.y = b1[48];
    c3 = __builtin_amdgcn_wmma_f32_16x16x4_f32(false, a, false, b, (short)0, c3, false, false);
  }

  const int mbase = half * 8;
  float* cptr = C + ((long long)tileM * 16 + mbase) * N + mn;
#pragma unroll
  for (int r = 0; r < 8; ++r) {
    float* row = cptr + (long long)r * N;
    row[0]  = c0[r];
    row[16] = c1[r];
    row[32] = c2[r];
    row[48] = c3[r];
  }
}

// ---------------- edge aggregation: out[dst] += h[src] * dinv[src]*dinv[dst] ----------------
// 16 lanes per edge, float4 per lane (64 feats). Gathers + f32 atomics hit L2
// (h/out = 25.6MB each, resident in the 192MB L2).
__global__ void __launch_bounds__(256)
agg_kernel(const float* __restrict__ h, const long long* __restrict__ src,
           const long long* __restrict__ dst, const float* __restrict__ dinv,
           float* __restrict__ out, long long E) {
  long long tid = (long long)blockIdx.x * blockDim.x + threadIdx.x;
  long long e = tid >> 4;
  if (e >= E) return;
  int f4 = (int)(tid & 15) << 2;
  int s = (int)src[e];
  int d = (int)dst[e];
  float norm = dinv[s] * dinv[d];
  float4 hv = *reinterpret_cast<const float4*>(h + (long long)s * 64 + f4);
  float* o = out + (long long)d * 64 + f4;
  atomicAdd(o + 0, hv.x * norm);
  atomicAdd(o + 1, hv.y * norm);
  atomicAdd(o + 2, hv.z * norm);
  atomicAdd(o + 3, hv.w * norm);
}

// ---------------- h = relu(agg + h*dinv^2 + b)  (writes over h) ----------------
__global__ void post1_kernel(float* __restrict__ h, const float* __restrict__ agg,
                             const float* __restrict__ bias,
                             const float* __restrict__ dinv, long long total) {
  long long t = (long long)blockIdx.x * blockDim.x + threadIdx.x;
  if (t >= total) return;
  long long node = t >> 6;
  int f = (int)(t & 63);
  float di = dinv[node];
  float v = agg[t] + h[t] * di * di + bias[f];
  h[t] = fmaxf(v, 0.0f);
}

// ---------------- fused layer-2 activation + column-sum readout ----------------
__global__ void __launch_bounds__(256)
post2_readout_kernel(const float* __restrict__ agg, const float* __restrict__ hpre,
                     const float* __restrict__ dinv, const float* __restrict__ bias,
                     float* __restrict__ gsum, int N) {
  __shared__ float partial[64];
  if (threadIdx.x < 64) partial[threadIdx.x] = 0.0f;
  __syncthreads();
  int f = threadIdx.x & 63;
  int sub = threadIdx.x >> 6;  // 0..3
  float acc = 0.0f;
  for (long long node = (long long)blockIdx.x * 4 + sub; node < N;
       node += (long long)gridDim.x * 4) {
    float di = dinv[node];
    long long idx = node * 64 + f;
    float v = agg[idx] + hpre[idx] * di * di + bias[f];
    acc += fmaxf(v, 0.0f);
  }
  atomicAdd(&partial[f], acc);
  __syncthreads();
  if (threadIdx.x < 64) atomicAdd(&gsum[threadIdx.x], partial[threadIdx.x]);
}

// ---------------- final: sigmoid(mean(h2) @ Wf + bf) ----------------
__global__ void final_kernel(const float* __restrict__ gsum, const float* __restrict__ Wf,
                             const float* __restrict__ bf, float* __restrict__ out,
                             float invN) {
  int lane = threadIdx.x;  // 32 threads, wave32
  float v = gsum[lane] * invN * Wf[lane] + gsum[lane + 32] * invN * Wf[lane + 32];
#pragma unroll
  for (int off = 16; off > 0; off >>= 1) v += __shfl_xor(v, off, 32);
  if (lane == 0) out[0] = 1.0f / (1.0f + expf(-(v + bf[0])));
}

// ---------------- launcher ----------------

extern "C" void kernel_launch(void* const* d_in, const int* in_sizes, int n_in,
                              void* d_out, int out_size, void* d_ws, size_t ws_size,
                              hipStream_t stream) {
  (void)n_in; (void)out_size; (void)ws_size;
  const float*     x  = (const float*)d_in[0];
  const long long* ei = (const long long*)d_in[1];
  const float*     W1 = (const float*)d_in[2];
  const float*     b1 = (const float*)d_in[3];
  const float*     W2 = (const float*)d_in[4];
  const float*     b2 = (const float*)d_in[5];
  const float*     Wf = (const float*)d_in[6];
  const float*     bf = (const float*)d_in[7];
  float* out = (float*)d_out;

  const int IN_DIM = 256, HID = 64;
  const int N = in_sizes[0] / IN_DIM;              // 100000
  const long long E = (long long)in_sizes[1] / 2;  // 3200000
  const long long* src = ei;
  const long long* dst = ei + E;

  char* ws = (char*)d_ws;
  size_t NB = (size_t)N * HID * sizeof(float);     // 25.6 MB
  float*    bufA = (float*)(ws);
  float*    bufB = (float*)(ws + NB);
  unsigned* deg  = (unsigned*)(ws + 2 * NB);
  float*    dinv = (float*)(ws + 2 * NB + (size_t)N * 4);
  float*    gsum = (float*)(ws + 2 * NB + (size_t)N * 8);

  // symmetric-normalization coefficients
  hipMemsetAsync(deg, 0, (size_t)N * 4, stream);
  deg_count_kernel<<<(unsigned)((E + 255) / 256), 256, 0, stream>>>(dst, deg, E);
  dinv_kernel<<<(unsigned)((N + 255) / 256), 256, 0, stream>>>(deg, dinv, N);

  const int tilesM = (N + 15) / 16;                         // 6250
  const unsigned gemm_blocks = (unsigned)((tilesM + 7) / 8); // 8 waves/block, 1 row-tile/wave
  const unsigned agg_blocks  = (unsigned)((E * 16 + 255) / 256);
  const long long total = (long long)N * HID;
  const unsigned ew_blocks = (unsigned)((total + 255) / 256);

  // layer 1: h1 = x@W1 -> bufA ; agg -> bufB ; bufA = relu(bufB + bufA*dinv^2 + b1)
  gemm_wmma_f32<<<gemm_blocks, 256, 0, stream>>>(x, W1, bufA, N, IN_DIM, HID);
  hipMemsetAsync(bufB, 0, NB, stream);
  agg_kernel<<<agg_blocks, 256, 0, stream>>>(bufA, src, dst, dinv, bufB, E);
  post1_kernel<<<ew_blocks, 256, 0, stream>>>(bufA, bufB, b1, dinv, total);

  // layer 2: h2pre = bufA@W2 -> bufB ; agg -> bufA ; fused relu + readout
  gemm_wmma_f32<<<gemm_blocks, 256, 0, stream>>>(bufA, W2, bufB, N, HID, HID);
  hipMemsetAsync(bufA, 0, NB, stream);
  agg_kernel<<<agg_blocks, 256, 0, stream>>>(bufB, src, dst, dinv, bufA, E);

  hipMemsetAsync(gsum, 0, HID * sizeof(float), stream);
  post2_readout_kernel<<<4096, 256, 0, stream>>>(bufA, bufB, dinv, b2, gsum, N);
  final_kernel<<<1, 32, 0, stream>>>(gsum, Wf, bf, out, 1.0f / (float)N);
}